// GCNModel_2147483648540
// MI455X (gfx1250) — compile-verified
//
#include <hip/hip_runtime.h>
#include <hip/hip_bf16.h>
#include <math.h>

typedef __attribute__((ext_vector_type(2))) float v2f;
typedef __attribute__((ext_vector_type(8))) float v8f;

#define NN     100000
#define FIN    512
#define FHID   128
#define FOUTD  40
#define MTILES (NN / 16)      // 6250 exactly
#define WPB    10             // waves per block (6250 = 625 * 10, no idle waves)
#define GBLK   (MTILES / WPB) // 625 blocks
#define BSTR1  288            // GEMM1 LDS pair-row stride (2*128 + 32): bank-disjoint halves
#define BSTR2  96             // GEMM2 LDS pair-row stride (2*48): 96%64==32 -> disjoint halves

// ---------------- degree / norm ----------------
__global__ void deg_init_kernel(float* __restrict__ deg, int n) {
    int i = blockIdx.x * blockDim.x + threadIdx.x;
    if (i < n) deg[i] = 1.0f;                    // self-loop
}

__global__ void deg_count_kernel(const int* __restrict__ dst, float* __restrict__ deg, int E) {
    int i = blockIdx.x * blockDim.x + threadIdx.x;
    if (i < E) atomicAdd(&deg[dst[i]], 1.0f);
}

__global__ void deg_rsqrt_kernel(float* __restrict__ deg, int n) {
    int i = blockIdx.x * blockDim.x + threadIdx.x;
    if (i < n) deg[i] = rsqrtf(deg[i]);
}

// ---------------- GEMM1: [NN,512] x [512,128] -> [NN,128] (fp32 WMMA, W1 via LDS) ---
// LDS holds one 128-row K-chunk of W1, pair-interleaved: element (k,n) at
// Bs[(k>>1)*BSTR1 + 2n + (k&1)], so the B fragment (K,N),(K+1,N) is one ds_load_b64.
__global__ void __launch_bounds__(32 * WPB) gemm1_kernel(const float* __restrict__ A,
                                                         const float* __restrict__ B,
                                                         float* __restrict__ C) {
    __shared__ float Bs[64 * BSTR1];             // 72 KB
    const int tid  = threadIdx.x;
    const int lane = tid & 31;
    const int tile = blockIdx.x * WPB + (tid >> 5);   // always < MTILES
    const int m0   = tile * 16;
    const int half = lane >> 4;                  // 0 or 1
    const int l16  = lane & 15;
    const int kOfs = half * 2;                   // K sub-offset per ISA A/B layout
    const int aRow = m0 + l16;
    const int bOff = half * BSTR1 + 2 * l16;     // per-lane LDS offset within pair-row

    v8f acc[8];
#pragma unroll
    for (int t = 0; t < 8; ++t)
#pragma unroll
        for (int j = 0; j < 8; ++j) acc[t][j] = 0.0f;

    const float* aPtr = A + (size_t)aRow * FIN + kOfs;
    for (int c = 0; c < FIN / 128; ++c) {
        // cooperative stage of W1 rows [c*128, c*128+128) -> LDS (pair-interleaved)
        for (int g = tid; g < 64 * FHID; g += 32 * WPB) {
            int p = g >> 7;                      // K-pair within chunk
            int n = g & 127;                     // column
            const float* sp = B + (size_t)(c * 128 + 2 * p) * FHID + n;
            v2f v;
            v.x = sp[0];
            v.y = sp[FHID];
            *(v2f*)&Bs[p * BSTR1 + 2 * n] = v;
        }
        __syncthreads();

        for (int kl = 0; kl < 128; kl += 4) {
            v2f a = *(const v2f*)(aPtr + c * 128 + kl);   // A frag (K, K+1)
            const float* bBase = &Bs[(kl >> 1) * BSTR1 + bOff];
#pragma unroll
            for (int t = 0; t < 8; ++t) {
                v2f b = *(const v2f*)(bBase + 32 * t);    // single ds_load_b64
                acc[t] = __builtin_amdgcn_wmma_f32_16x16x4_f32(
                    false, a, false, b, (short)0, acc[t], false, false);
            }
        }
        __syncthreads();
    }
    const int rowBase = m0 + half * 8;           // C/D layout: M = j + 8*half
#pragma unroll
    for (int t = 0; t < 8; ++t)
#pragma unroll
        for (int j = 0; j < 8; ++j)
            C[(size_t)(rowBase + j) * FHID + t * 16 + l16] = acc[t][j];
}

// ---------------- GEMM2: [NN,128] x [128,40] -> [NN,40] (fp32 WMMA, W2 via LDS) -----
// W2 pair-interleaved with columns zero-padded 40 -> 48, so the inner loop is
// unguarded ds_load_b64; only the stores mask col >= 40.
__global__ void __launch_bounds__(32 * WPB) gemm2_kernel(const float* __restrict__ A,
                                                         const float* __restrict__ B,
                                                         float* __restrict__ C) {
    __shared__ float W2s[64 * BSTR2];            // 24 KB
    const int tid  = threadIdx.x;
    const int lane = tid & 31;
    for (int g = tid; g < 64 * BSTR2; g += 32 * WPB) {
        int p   = g / BSTR2;                     // K-pair
        int rem = g % BSTR2;
        int n   = rem >> 1;                      // padded column (0..47)
        int lo  = rem & 1;
        int k   = 2 * p + lo;
        W2s[g] = (n < FOUTD) ? B[(size_t)k * FOUTD + n] : 0.0f;
    }
    __syncthreads();

    const int tile = blockIdx.x * WPB + (tid >> 5);
    const int m0   = tile * 16;
    const int half = lane >> 4;
    const int l16  = lane & 15;
    const int kOfs = half * 2;
    const int aRow = m0 + l16;
    const int bOff = half * BSTR2 + 2 * l16;

    v8f acc[3];
#pragma unroll
    for (int t = 0; t < 3; ++t)
#pragma unroll
        for (int j = 0; j < 8; ++j) acc[t][j] = 0.0f;

    const float* aPtr = A + (size_t)aRow * FHID + kOfs;
    for (int k0 = 0; k0 < FHID; k0 += 4) {
        v2f a = *(const v2f*)(aPtr + k0);
        const float* bBase = &W2s[(k0 >> 1) * BSTR2 + bOff];
#pragma unroll
        for (int t = 0; t < 3; ++t) {
            v2f b = *(const v2f*)(bBase + 32 * t);
            acc[t] = __builtin_amdgcn_wmma_f32_16x16x4_f32(
                false, a, false, b, (short)0, acc[t], false, false);
        }
    }
    const int rowBase = m0 + half * 8;
#pragma unroll
    for (int t = 0; t < 3; ++t) {
        int col = t * 16 + l16;
        if (col < FOUTD) {
#pragma unroll
            for (int j = 0; j < 8; ++j)
                C[(size_t)(rowBase + j) * FOUTD + col] = acc[t][j];
        }
    }
}

// ---------------- self-loop term: out = h * dinv^2 ----------------
template <int F>
__global__ void selfloop_kernel(const float* __restrict__ h, const float* __restrict__ dinv,
                                float* __restrict__ out, long long total) {
    long long idx = (long long)blockIdx.x * blockDim.x + threadIdx.x;
    if (idx >= total) return;
    int i = (int)(idx / F);
    float di = dinv[i];
    out[idx] = h[idx] * di * di;
}

// ---------------- edge scatter layer 1: one wave per edge, 128 feats ----------------
__global__ void scatter1_kernel(const int* __restrict__ src, const int* __restrict__ dst,
                                const float* __restrict__ h, const float* __restrict__ dinv,
                                float* __restrict__ out, int E) {
    int idx  = blockIdx.x * blockDim.x + threadIdx.x;
    int e    = idx >> 5;
    int lane = idx & 31;
    if (e >= E) return;
    int s = src[e], d = dst[e];
    float norm = dinv[s] * dinv[d];
    const float* hp = h + (size_t)s * FHID;
    float*       op = out + (size_t)d * FHID;
#pragma unroll
    for (int i = 0; i < 4; ++i) {
        int f = lane + i * 32;
        atomicAdd(op + f, hp[f] * norm);
    }
}

// ---------------- bias + relu in place (layer 1) ----------------
__global__ void bias_relu_kernel(float* __restrict__ a, const float* __restrict__ b, long long total) {
    long long idx = (long long)blockIdx.x * blockDim.x + threadIdx.x;
    if (idx >= total) return;
    int f = (int)(idx & (FHID - 1));
    a[idx] = fmaxf(a[idx] + b[f], 0.0f);
}

// ---------------- edge scatter layer 2: one thread per (edge, feat) ----------------
__global__ void scatter2_kernel(const int* __restrict__ src, const int* __restrict__ dst,
                                const float* __restrict__ h, const float* __restrict__ dinv,
                                float* __restrict__ out, int E) {
    long long idx = (long long)blockIdx.x * blockDim.x + threadIdx.x;
    long long total = (long long)E * FOUTD;
    if (idx >= total) return;
    int e = (int)(idx / FOUTD);
    int f = (int)(idx % FOUTD);
    int s = src[e], d = dst[e];
    float norm = dinv[s] * dinv[d];
    atomicAdd(out + (size_t)d * FOUTD + f, h[(size_t)s * FOUTD + f] * norm);
}

// ---------------- +b2, log_softmax in place: one wave per row ----------------
__global__ void finalize_kernel(float* __restrict__ out, const float* __restrict__ b2) {
    int row  = blockIdx.x * (blockDim.x >> 5) + (threadIdx.x >> 5);
    int lane = threadIdx.x & 31;
    if (row >= NN) return;
    float* p = out + (size_t)row * FOUTD;
    float a  = p[lane] + b2[lane];                                    // cols 0..31
    float bv = (lane < 8) ? (p[32 + lane] + b2[32 + lane]) : -1e30f;  // cols 32..39
    float m = fmaxf(a, bv);
    for (int off = 16; off; off >>= 1) m = fmaxf(m, __shfl_xor(m, off, 32));
    float ea = expf(a - m);
    float eb = (lane < 8) ? expf(bv - m) : 0.0f;
    float s = ea + eb;
    for (int off = 16; off; off >>= 1) s += __shfl_xor(s, off, 32);
    float lse = m + logf(s);
    p[lane] = a - lse;
    if (lane < 8) p[32 + lane] = bv - lse;
}

extern "C" void kernel_launch(void* const* d_in, const int* in_sizes, int n_in,
                              void* d_out, int out_size, void* d_ws, size_t ws_size,
                              hipStream_t stream) {
    const float* x  = (const float*)d_in[0];
    const int*   ei = (const int*)d_in[1];
    const float* W1 = (const float*)d_in[2];
    const float* b1 = (const float*)d_in[3];
    const float* W2 = (const float*)d_in[4];
    const float* b2 = (const float*)d_in[5];
    float* out = (float*)d_out;

    const int E = in_sizes[1] / 2;
    const int* srcIdx = ei;
    const int* dstIdx = ei + E;

    char* wsb   = (char*)d_ws;
    float* dinv = (float*)wsb;                                   // 400 KB
    float* h1   = (float*)(wsb + (size_t)(512 * 1024));          // 51.2 MB (L2-resident)
    float* agg1 = (float*)(wsb + (size_t)(52) * 1024 * 1024);    // 51.2 MB (L2-resident)
    float* h2   = h1;                                            // reuse h1 after layer 1

    const int T = 256;
    // degrees with self-loops -> dinv
    deg_init_kernel<<<(NN + T - 1) / T, T, 0, stream>>>(dinv, NN);
    deg_count_kernel<<<(E + T - 1) / T, T, 0, stream>>>(dstIdx, dinv, E);
    deg_rsqrt_kernel<<<(NN + T - 1) / T, T, 0, stream>>>(dinv, NN);

    // layer 1
    gemm1_kernel<<<GBLK, 32 * WPB, 0, stream>>>(x, W1, h1);
    long long tot1 = (long long)NN * FHID;
    selfloop_kernel<FHID><<<(int)((tot1 + T - 1) / T), T, 0, stream>>>(h1, dinv, agg1, tot1);
    scatter1_kernel<<<(int)(((long long)E * 32 + T - 1) / T), T, 0, stream>>>(srcIdx, dstIdx, h1, dinv, agg1, E);
    bias_relu_kernel<<<(int)((tot1 + T - 1) / T), T, 0, stream>>>(agg1, b1, tot1);

    // layer 2
    gemm2_kernel<<<GBLK, 32 * WPB, 0, stream>>>(agg1, W2, h2);
    long long tot2 = (long long)NN * FOUTD;
    selfloop_kernel<FOUTD><<<(int)((tot2 + T - 1) / T), T, 0, stream>>>(h2, dinv, out, tot2);
    scatter2_kernel<<<(int)(((long long)E * FOUTD + T - 1) / T), T, 0, stream>>>(srcIdx, dstIdx, h2, dinv, out, E);

    // bias + log_softmax
    finalize_kernel<<<(NN + 7) / 8, T, 0, stream>>>(out, b2);
}